// PitLoss_5506148073560
// MI455X (gfx1250) — compile-verified
//
#include <hip/hip_runtime.h>
#include <math.h>

typedef __attribute__((ext_vector_type(2))) float v2f;
typedef __attribute__((ext_vector_type(8))) float v8f;

#define TPB 256
#define PIT_EPS 1e-14f

// One block per batch element. Streams T frames, accumulates per-lane:
//   H[i]   = sum_t mask * (max(x_i,0)+log1p(exp(-|x_i|)))   (3 vals)
//   G[i][j]= sum_t (mask*x_i) * targ_j                       (9 vals)
//   ms     = sum_t mask                                      (1 val)
// then block-reduces 13 columns with chained V_WMMA_F32_16X16X4_F32
// (A = ones -> column sums), and finally does the 6-permutation min/argmin.
__global__ __launch_bounds__(TPB) void pit_batch_kernel(
    const float* __restrict__ pred,
    const float* __restrict__ targ,
    const float* __restrict__ mask,
    float* __restrict__ out,        // out[0]=mean (written by finalize), out_int[1+b]=argmin
    float* __restrict__ ws_best,    // per-batch best value
    int T)
{
    const int b   = blockIdx.x;
    const int tid = threadIdx.x;
    const size_t baseBT = (size_t)b * (size_t)T;

    float H[3]    = {0.f, 0.f, 0.f};
    float G[3][3] = {{0.f,0.f,0.f},{0.f,0.f,0.f},{0.f,0.f,0.f}};
    float ms = 0.f;

    for (int t = tid; t < T; t += TPB) {
        const size_t ft = baseBT + (size_t)t;
        const size_t p3 = ft * 3;

        // Streaming data used exactly once: non-temporal (dataset > 192MB L2).
        float m  = __builtin_nontemporal_load(mask + ft);
        float x0 = __builtin_nontemporal_load(pred + p3 + 0);
        float x1 = __builtin_nontemporal_load(pred + p3 + 1);
        float x2 = __builtin_nontemporal_load(pred + p3 + 2);
        float y0 = __builtin_nontemporal_load(targ + p3 + 0);
        float y1 = __builtin_nontemporal_load(targ + p3 + 1);
        float y2 = __builtin_nontemporal_load(targ + p3 + 2);

        // Prefetch next tile (global_prefetch_b8).
        if (t + TPB < T) {
            __builtin_prefetch(pred + p3 + (size_t)TPB * 3, 0, 0);
            __builtin_prefetch(targ + p3 + (size_t)TPB * 3, 0, 0);
        }

        float x[3] = {x0, x1, x2};
        float y[3] = {y0, y1, y2};
        float xm[3];
#pragma unroll
        for (int i = 0; i < 3; ++i) {
            float h = fmaxf(x[i], 0.f) + log1pf(__expf(-fabsf(x[i])));
            H[i]  = fmaf(m, h, H[i]);
            xm[i] = m * x[i];
        }
#pragma unroll
        for (int i = 0; i < 3; ++i)
#pragma unroll
            for (int j = 0; j < 3; ++j)
                G[i][j] = fmaf(xm[i], y[j], G[i][j]);
        ms += m;
    }

    // ---- Block reduction of 13 accumulators via WMMA column sums ----
    __shared__ float red[TPB * 16];   // 16 KB (of 320 KB/WGP)
    __shared__ float colsum[16];

    {
        float acc[16];
        acc[0] = H[0]; acc[1] = H[1]; acc[2] = H[2];
#pragma unroll
        for (int i = 0; i < 3; ++i)
#pragma unroll
            for (int j = 0; j < 3; ++j)
                acc[3 + i*3 + j] = G[i][j];
        acc[12] = ms; acc[13] = 0.f; acc[14] = 0.f; acc[15] = 0.f;
#pragma unroll
        for (int k = 0; k < 16; ++k)
            red[tid * 16 + k] = acc[k];
    }
    __syncthreads();

    if (tid < 32) {                    // whole wave 0: EXEC all-ones for WMMA
        v2f a = {1.0f, 1.0f};          // A = ones 16x4 -> D[m][n] = sum_k B[k][n]
        v8f c = {0.f,0.f,0.f,0.f,0.f,0.f,0.f,0.f};
        const int col = tid & 15;
        const int grp = tid >> 4;
        for (int it = 0; it < TPB / 4; ++it) {
            // B (4x16 f32, 2 VGPRs): each (lane-group, VGPR) pair supplies one
            // distinct K-row; with A=ones any bijection of rows is correct.
            v2f bm;
            bm[0] = red[(4 * it + grp    ) * 16 + col];
            bm[1] = red[(4 * it + grp + 2) * 16 + col];
            c = __builtin_amdgcn_wmma_f32_16x16x4_f32(
                    /*neg_a=*/false, a, /*neg_b=*/false, bm,
                    /*c_mod=*/(short)0, c, /*reuse_a=*/false, /*reuse_b=*/false);
        }
        // D row M=0 lives in c[0] of lanes 0..15, N = lane.
        if (tid < 16) colsum[tid] = c[0];
    }
    __syncthreads();

    if (tid == 0) {
        float inv = 1.f / (colsum[12] + PIT_EPS);
        float cost[3][3];
#pragma unroll
        for (int i = 0; i < 3; ++i)
#pragma unroll
            for (int j = 0; j < 3; ++j)
                cost[i][j] = (colsum[i] - colsum[3 + i*3 + j]) * inv;

        // itertools.permutations(range(3)) order:
        const int P[6][3] = {{0,1,2},{0,2,1},{1,0,2},{1,2,0},{2,0,1},{2,1,0}};
        float best = 0.f; int bi = 0;
#pragma unroll
        for (int p = 0; p < 6; ++p) {
            float l = (cost[0][P[p][0]] + cost[1][P[p][1]] + cost[2][P[p][2]])
                      * (1.0f / 3.0f);
            if (p == 0 || l < best) { best = l; bi = p; }   // first-occurrence argmin
        }
        ws_best[b] = best;
        reinterpret_cast<int*>(out)[1 + b] = bi;            // int32 best_idx
    }
}

// Deterministic mean of the 512 per-batch best values.
__global__ __launch_bounds__(512) void pit_final_kernel(
    const float* __restrict__ ws_best, float* __restrict__ out, int B)
{
    __shared__ float s[512];
    const int tid = threadIdx.x;
    s[tid] = (tid < B) ? ws_best[tid] : 0.f;
    __syncthreads();
#pragma unroll
    for (int off = 256; off > 0; off >>= 1) {
        if (tid < off) s[tid] += s[tid + off];
        __syncthreads();
    }
    if (tid == 0) out[0] = s[0] / (float)B;
}

extern "C" void kernel_launch(void* const* d_in, const int* in_sizes, int n_in,
                              void* d_out, int out_size, void* d_ws, size_t ws_size,
                              hipStream_t stream) {
    const float* pred = (const float*)d_in[0];
    const float* targ = (const float*)d_in[1];
    const float* mask = (const float*)d_in[2];
    float* out = (float*)d_out;
    float* ws  = (float*)d_ws;

    const int B = out_size - 1;               // 512 (scalar + B indices)
    const int T = in_sizes[2] / B;            // mask has B*T elements -> 8000

    pit_batch_kernel<<<B, TPB, 0, stream>>>(pred, targ, mask, out, ws, T);
    pit_final_kernel<<<1, 512, 0, stream>>>(ws, out, B);
}